// StatefulSynapseNet_19662360281544
// MI455X (gfx1250) — compile-verified
//
#include <hip/hip_runtime.h>
#include <hip/hip_bf16.h>

// CDNA5 / gfx1250 wave32 SNN forward.
// Exact-fp32 path via V_WMMA_F32_16X16X4_F32 (D = A(16x4) * B(4x16) + C).
//
// Layout assumptions (CDNA5 ISA §7.12.2):
//  A 16x4 f32 : lanes 0-15 -> M=lane   {v0:K=0, v1:K=1}
//               lanes 16-31 -> M=lane-16 {v0:K=2, v1:K=3}
//  B 4x16 f32 : lanes 0-15 -> N=lane   {v0:K=0, v1:K=1}
//               lanes 16-31 -> N=lane-16 {v0:K=2, v1:K=3}
//  C/D 16x16  : lane&15 -> N, vgpr r -> M=r (+8 for lanes 16-31)

typedef __attribute__((ext_vector_type(2))) float v2f;
typedef __attribute__((ext_vector_type(8))) float v8f;

#define T_STEPS 28
#define F_IN    28
#define C1      32
#define C2      10
#define WAVES_PER_BLOCK 4
#define SAMPLES_PER_BLOCK (16 * WAVES_PER_BLOCK)

__global__ __launch_bounds__(32 * WAVES_PER_BLOCK)
void snn_fused_kernel(const float* __restrict__ x,     // [N, 28, 28] (n, f, t)
                      const float* __restrict__ W1,    // [32, 28]
                      const float* __restrict__ b1,    // [32]
                      const float* __restrict__ w_syn, // [1]
                      const float* __restrict__ W2,    // [10, 32]
                      const float* __restrict__ b2,    // [10]
                      float* __restrict__ out)         // [N, 10]
{
    // per-wave transpose buffer: y (D-layout) -> A-layout for GEMM2
    __shared__ float ylds[WAVES_PER_BLOCK][16][C1 + 1];

    const int lane = threadIdx.x & 31;
    const int wave = threadIdx.x >> 5;
    const int col  = lane & 15;   // tile column (N for B/C/D, M for A)
    const int hi   = lane >> 4;   // lane-half selects K pair / sample-row half
    const int koff = hi * 2;

    const int n0 = (blockIdx.x * WAVES_PER_BLOCK + wave) * 16;

    // ---- preload W1 / W2 into B-matrix register tiles ----
    v2f b1v[2][7];
#pragma unroll
    for (int ot = 0; ot < 2; ++ot) {
#pragma unroll
        for (int ks = 0; ks < 7; ++ks) {
            const int o = ot * 16 + col;
            const int f = ks * 4 + koff;
            b1v[ot][ks].x = W1[o * F_IN + f];
            b1v[ot][ks].y = W1[o * F_IN + f + 1];
        }
    }
    v2f b2v[8];
#pragma unroll
    for (int ks = 0; ks < 8; ++ks) {
        const int c = ks * 4 + koff;
        b2v[ks].x = (col < C2) ? W2[col * C1 + c]     : 0.0f;
        b2v[ks].y = (col < C2) ? W2[col * C1 + c + 1] : 0.0f;
    }
    const float bias1a = b1[col];
    const float bias1b = b1[16 + col];
    const float bias2  = (col < C2) ? b2[col] : 0.0f;
    const float inv_tau = 1.0f / (1.0f + __expf(-w_syn[0]));
    const float keep    = 1.0f - inv_tau;

    // neuron / synapse state in C/D layout
    v8f v1a = {}, v1b = {}, ga = {}, gb = {}, v2s = {}, acc = {};

    const float* xn = x + (size_t)(n0 + col) * (F_IN * T_STEPS);

    for (int t = 0; t < T_STEPS; ++t) {
        // ---- A1 tile: x[n0+col, f, t] ----
        v2f a1[7];
#pragma unroll
        for (int ks = 0; ks < 7; ++ks) {
            const int f = ks * 4 + koff;
            a1[ks].x = xn[f * T_STEPS + t];
            a1[ks].y = xn[(f + 1) * T_STEPS + t];
        }

        // ---- GEMM1: h[16 samples x 32 ch] = x_t * W1^T + b1 ----
        v8f d1a, d1b;
#pragma unroll
        for (int i = 0; i < 8; ++i) { d1a[i] = bias1a; d1b[i] = bias1b; }
#pragma unroll
        for (int ks = 0; ks < 7; ++ks)
            d1a = __builtin_amdgcn_wmma_f32_16x16x4_f32(
                false, a1[ks], false, b1v[0][ks], (short)0, d1a, false, false);
#pragma unroll
        for (int ks = 0; ks < 7; ++ks)
            d1b = __builtin_amdgcn_wmma_f32_16x16x4_f32(
                false, a1[ks], false, b1v[1][ks], (short)0, d1b, false, false);

        // ---- IF neuron 1 (hard reset) + synapse filter; park y in LDS ----
#pragma unroll
        for (int r = 0; r < 8; ++r) {
            const int m = r + hi * 8;

            float v = v1a[r] + d1a[r];
            float s = (v >= 1.0f) ? 1.0f : 0.0f;
            v1a[r]  = (v >= 1.0f) ? 0.0f : v;
            float g = ga[r] * keep + s;
            ga[r]   = g;
            ylds[wave][m][col] = g;

            v = v1b[r] + d1b[r];
            s = (v >= 1.0f) ? 1.0f : 0.0f;
            v1b[r] = (v >= 1.0f) ? 0.0f : v;
            g = gb[r] * keep + s;
            gb[r] = g;
            ylds[wave][m][16 + col] = g;
        }
        __syncthreads();

        // ---- A2 tile: y[16 samples x 32 ch] from LDS ----
        v2f a2[8];
#pragma unroll
        for (int ks = 0; ks < 8; ++ks) {
            const int c = ks * 4 + koff;
            a2[ks].x = ylds[wave][col][c];
            a2[ks].y = ylds[wave][col][c + 1];
        }
        __syncthreads();

        // ---- GEMM2: h2[16 x 16(pad of 10)] = y * W2^T + b2 ----
        v8f d2;
#pragma unroll
        for (int i = 0; i < 8; ++i) d2[i] = bias2;
#pragma unroll
        for (int ks = 0; ks < 8; ++ks)
            d2 = __builtin_amdgcn_wmma_f32_16x16x4_f32(
                false, a2[ks], false, b2v[ks], (short)0, d2, false, false);

        // ---- IF neuron 2 + spike accumulation ----
#pragma unroll
        for (int r = 0; r < 8; ++r) {
            float v = v2s[r] + d2[r];
            float s = (v >= 1.0f) ? 1.0f : 0.0f;
            v2s[r]  = (v >= 1.0f) ? 0.0f : v;
            acc[r] += s;
        }
    }

    // ---- out[n, o] = mean_t s2 ----
    if (col < C2) {
#pragma unroll
        for (int r = 0; r < 8; ++r) {
            const int m = r + hi * 8;
            out[(size_t)(n0 + m) * C2 + col] = acc[r] * (1.0f / T_STEPS);
        }
    }
}

extern "C" void kernel_launch(void* const* d_in, const int* in_sizes, int n_in,
                              void* d_out, int out_size, void* d_ws, size_t ws_size,
                              hipStream_t stream) {
    const float* x    = (const float*)d_in[0];
    const float* W1   = (const float*)d_in[1];
    const float* b1   = (const float*)d_in[2];
    const float* wsyn = (const float*)d_in[3];
    const float* W2   = (const float*)d_in[4];
    const float* b2   = (const float*)d_in[5];
    float* out = (float*)d_out;

    const int N = in_sizes[0] / (F_IN * T_STEPS);   // 65536
    const int blocks = (N + SAMPLES_PER_BLOCK - 1) / SAMPLES_PER_BLOCK;

    snn_fused_kernel<<<blocks, 32 * WAVES_PER_BLOCK, 0, stream>>>(
        x, W1, b1, wsyn, W2, b2, out);
}